// MGL_14628658610800
// MI455X (gfx1250) — compile-verified
//
#include <hip/hip_runtime.h>

typedef __attribute__((ext_vector_type(2))) float v2f;
typedef __attribute__((ext_vector_type(8))) float v8f;
typedef __attribute__((ext_vector_type(4))) unsigned int u32x4;
typedef __attribute__((ext_vector_type(8))) int i32x8;
typedef __attribute__((ext_vector_type(4))) int i32x4;

#define HH   96
#define WW   96
#define HWP  9216
#define CC   64
#define TT   4
#define NIMG 8                       // n*t images
#define YSZ  (NIMG * CC * HWP)       // floats per conv-output buffer (4,718,592)

__device__ __forceinline__ v8f wmma4(v2f a, v2f b, v8f c) {
  // D(16x16,f32) = A(16x4,f32) * B(4x16,f32) + C
  return __builtin_amdgcn_wmma_f32_16x16x4_f32(
      /*neg_a=*/false, a, /*neg_b=*/false, b,
      /*c_mod=*/(short)0, c, /*reuse_a=*/false, /*reuse_b=*/false);
}

// Low 32 bits of a generic pointer to LDS == LDS byte offset on gfx12xx.
__device__ __forceinline__ unsigned lds_addr32(const void* p) {
  return (unsigned)(unsigned long long)(uintptr_t)p;
}

// ---------------------------------------------------------------------------
// Kernel A: y_i = relu(W_i * X) for i=1..3, per image. One 64(M)x32(N) tile
// per 256-thread block (8 waves, each a 16x16 WMMA tile), K=64 via 16 x
// v_wmma_f32_16x16x4_f32 per weight set. X tile is fetched by the Tensor
// Data Mover (2D D#: 32 floats/row, 64 rows, stride 9216) while the waves
// stage the weights.
// ---------------------------------------------------------------------------
__global__ __launch_bounds__(256) void conv1x1_relu3(
    const float* __restrict__ x, const float* __restrict__ w1,
    const float* __restrict__ w2, const float* __restrict__ w3,
    float* __restrict__ y) {
  __shared__ float sW[3][64][64];   // 48 KB
  __shared__ float sX[64][32];      // 8 KB, filled by TDM
  const int img = blockIdx.y;       // 0..7  (n*4 + t)
  const int p0  = blockIdx.x * 32;  // pixel base
  const int tid = threadIdx.x;

  // ---- X tile via TDM (wave 0 issues; TENSORcnt is per-wave) ----
  if (tid < 32) {
    const unsigned lds = lds_addr32(&sX[0][0]);
    const unsigned long long ga =
        (unsigned long long)(uintptr_t)(x + (size_t)img * (CC * HWP) + p0);
    u32x4 g0 = {
        1u,                                   // count=1, user mode, no gather
        lds,                                  // lds_addr
        (unsigned)(ga & 0xffffffffu),         // global_addr[31:0]
        (unsigned)((ga >> 32) & 0x01ffffffu)  // global_addr[56:32]
            | (2u << 30)                      // type=2 ("image")
    };
    i32x8 g1 = {
        (int)(2u << 16),                      // wg_mask=0, data_size=2 (4B)
        (int)((HWP & 0xffffu) << 16),         // tensor_dim0[15:0]=9216
        (int)(((HWP >> 16) & 0xffffu) | (64u << 16)),  // t_dim0 hi | tensor_dim1=64
        (int)(32u << 16),                     // tensor_dim1 hi=0 | tile_dim0=32
        (int)64,                              // tile_dim1=64, tile_dim2=0
        (int)HWP,                             // tensor_dim0_stride[31:0]=9216
        0, 0                                  // stride0 hi, dim1_stride (unused, 2D)
    };
    i32x4 gz4 = {0, 0, 0, 0};
    i32x8 gz8 = {0, 0, 0, 0, 0, 0, 0, 0};
    __builtin_amdgcn_tensor_load_to_lds(g0, g1, gz4, gz4, gz8, 0);
    __builtin_amdgcn_s_wait_tensorcnt(0);
  }

  // ---- weights -> LDS (small, L2-resident broadcast) ----
  const float* wsrc[3] = {w1, w2, w3};
  for (int m = 0; m < 3; ++m)
    for (int i = tid; i < 64 * 64; i += 256)
      sW[m][i >> 6][i & 63] = wsrc[m][i];
  __syncthreads();

  const int lane = tid & 31;
  const int wv   = tid >> 5;              // 0..7
  const int mt   = wv & 3, nt = wv >> 2;  // mt 0..3, nt 0..1
  const int l15  = lane & 15;
  const int koff = (lane < 16) ? 0 : 2;   // ISA 32-bit A layout: half-wave K split
  const int moff = (lane < 16) ? 0 : 8;   // C/D row split
  const int mrow = mt * 16 + l15;
  const int ncol = nt * 16 + l15;

  for (int m = 0; m < 3; ++m) {
    v8f acc = {};
    for (int kb = 0; kb < 64; kb += 4) {
      v2f a, b;
      a.x = sW[m][mrow][kb + koff];
      a.y = sW[m][mrow][kb + koff + 1];
      b.x = sX[kb + koff][ncol];
      b.y = sX[kb + koff + 1][ncol];
      acc = wmma4(a, b, acc);
    }
    float* yb = y + (size_t)m * YSZ + (size_t)img * (CC * HWP);
    for (int v = 0; v < 8; ++v)
      yb[(size_t)(mt * 16 + v + moff) * HWP + (p0 + ncol)] = fmaxf(acc[v], 0.0f);
  }
}

// ---------------------------------------------------------------------------
// Kernel B: per-pixel attention. One pixel per 128-thread block (4 waves).
// The unfolded 3x3 window gather goes straight to LDS with per-lane
// GLOBAL_LOAD_ASYNC_TO_LDS_B32 (EXEC-predicated for the zero-pad boundary),
// synchronized on ASYNCcnt. score1 (48x48, K=64) and x5 = score1*X3 + x4
// (48x64, K=48 with exact zero padding) run on WMMA; x4 rides in as the WMMA
// C operand. Small t x t pieces on VALU.
// ---------------------------------------------------------------------------
__global__ __launch_bounds__(128) void mgl_attn(
    const float* __restrict__ y, float* __restrict__ out,
    const int* __restrict__ dil) {
  __shared__ float X1[48][64];          // u1, rows 36..47 = 0
  __shared__ float U2[48][64];          // u2 (x2f transposed), rows 36..47 = 0
  __shared__ float X3[48][64];          // u3, rows 36..47 = 0
  __shared__ float S[48][48];           // score1 logits -> probabilities
  __shared__ float X6[48][64];          // x4 then x6
  __shared__ float x1m[4][64], x1v[4][64], x2m[4][64], x2v[4][64];
  __shared__ float L2[4][4], L3[4][4], s23[4][4];
  __shared__ float s4[36];

  const int d    = dil[0];
  const int nIdx = blockIdx.x / HWP;    // batch
  const int p    = blockIdx.x % HWP;
  const int py   = p / WW, px = p % WW;
  const int tid  = threadIdx.x;

  // ---- gather unfolded 3x3 windows of y1/y2/y3 for this pixel (async) ----
  {
    const float* srcs[3] = {y, y + (size_t)YSZ, y + 2 * (size_t)YSZ};
    float* dsts[3] = {&X1[0][0], &U2[0][0], &X3[0][0]};
    for (int m = 0; m < 3; ++m) {
      const float* yb = srcs[m] + (size_t)nIdx * TT * CC * HWP;
      float* dst = dsts[m];
      for (int e = tid; e < 48 * 64; e += 128) {
        int r = e >> 6, c = e & 63;
        bool valid = false;
        unsigned long long ga = 0;
        if (r < 36) {
          int t = r / 9, k = r % 9;
          int yy = py + (k / 3 - 1) * d;
          int xx = px + (k % 3 - 1) * d;
          if (yy >= 0 && yy < HH && xx >= 0 && xx < WW) {
            valid = true;
            ga = (unsigned long long)(uintptr_t)
                 (yb + (size_t)(t * CC + c) * HWP + yy * WW + xx);
          }
        }
        if (valid) {
          unsigned lds = lds_addr32(dst + e);
          asm volatile("global_load_async_to_lds_b32 %0, %1, off"
                       :: "v"(lds), "v"(ga) : "memory");
        } else {
          dst[e] = 0.0f;                 // zero padding / out-of-image taps
        }
      }
    }
    asm volatile("s_wait_asynccnt 0x0" ::: "memory");
  }
  __syncthreads();

  // ---- per-(t,c) mean / unbiased var over the 9 window taps ----
  for (int e = tid; e < 256; e += 128) {
    int t = e >> 6, c = e & 63;
    float s1 = 0.f, s2 = 0.f;
    for (int k = 0; k < 9; ++k) { s1 += X1[t * 9 + k][c]; s2 += U2[t * 9 + k][c]; }
    float m1 = s1 * (1.0f / 9.0f), m2 = s2 * (1.0f / 9.0f);
    float v1 = 0.f, v2 = 0.f;
    for (int k = 0; k < 9; ++k) {
      float a = X1[t * 9 + k][c] - m1; v1 += a * a;
      float b = U2[t * 9 + k][c] - m2; v2 += b * b;
    }
    x1m[t][c] = m1; x1v[t][c] = v1 * 0.125f;
    x2m[t][c] = m2; x2v[t][c] = v2 * 0.125f;
  }
  __syncthreads();

  // ---- score2/score3 logits (4x4, K=64) ----
  if (tid < 16) {
    int t = tid >> 2, u = tid & 3;
    float l2 = 0.f, l3 = 0.f;
    for (int c = 0; c < 64; ++c) {
      l2 += x1v[t][c] * x2v[u][c];
      l3 += x1m[t][c] * x2m[u][c];
    }
    L2[t][u] = l2; L3[t][u] = l3;
  }
  __syncthreads();
  if (tid < 4) {
    int t = tid;
    float m2 = L2[t][0], m3 = L3[t][0];
    for (int u = 1; u < 4; ++u) { m2 = fmaxf(m2, L2[t][u]); m3 = fmaxf(m3, L3[t][u]); }
    float e2[4], e3[4]; float z2 = 0.f, z3 = 0.f;
    for (int u = 0; u < 4; ++u) {
      e2[u] = __expf(L2[t][u] - m2); z2 += e2[u];
      e3[u] = __expf(L3[t][u] - m3); z3 += e3[u];
    }
    for (int u = 0; u < 4; ++u) s23[t][u] = e2[u] / z2 + e3[u] / z3;
  }

  // ---- score1 logits: S(48x48) = X1(48x64) * U2^T, 9 WMMA tiles / 4 waves ----
  const int lane = tid & 31;
  const int wv   = tid >> 5;
  const int l15  = lane & 15;
  const int koff = (lane < 16) ? 0 : 2;
  const int moff = (lane < 16) ? 0 : 8;

  for (int tile = wv; tile < 9; tile += 4) {
    const int mb = (tile / 3) * 16, nb = (tile % 3) * 16;
    v8f acc = {};
    for (int kb = 0; kb < 64; kb += 4) {
      v2f a, b;
      a.x = X1[mb + l15][kb + koff];
      a.y = X1[mb + l15][kb + koff + 1];
      b.x = U2[nb + l15][kb + koff];      // B[k][n] = x2f[c=k][j=n] = U2[n][k]
      b.y = U2[nb + l15][kb + koff + 1];
      acc = wmma4(a, b, acc);
    }
    for (int v = 0; v < 8; ++v) S[mb + v + moff][nb + l15] = acc[v];
  }
  __syncthreads();

  // ---- row softmax over the 36 valid columns; zero the K padding ----
  if (tid < 36) {
    const int r = tid;
    float mx = S[r][0];
    for (int j = 1; j < 36; ++j) mx = fmaxf(mx, S[r][j]);
    float z = 0.f;
    for (int j = 0; j < 36; ++j) { float e = __expf(S[r][j] - mx); S[r][j] = e; z += e; }
    float inv = 1.0f / z;
    for (int j = 0; j < 36; ++j) S[r][j] *= inv;
    for (int j = 36; j < 48; ++j) S[r][j] = 0.f;   // exact K padding for x5
  }
  __syncthreads();

  // ---- x4 = (score2+score3) * X3  ->  X6 (rows 36..47 zeroed) ----
  for (int e = tid; e < 48 * 64; e += 128) {
    int r = e >> 6, c = e & 63;
    float v = 0.f;
    if (r < 36) {
      int t = r / 9, k = r % 9;
      v = s23[t][0] * X3[k][c]      + s23[t][1] * X3[9 + k][c] +
          s23[t][2] * X3[18 + k][c] + s23[t][3] * X3[27 + k][c];
    }
    X6[r][c] = v;
  }
  __syncthreads();

  // ---- x6 = score1 * X3 + x4 : 12 WMMA tiles (48x64, K=48), C = x4 ----
  for (int tile = wv; tile < 12; tile += 4) {
    const int mb = (tile >> 2) * 16, nb = (tile & 3) * 16;
    v8f acc;
    for (int v = 0; v < 8; ++v) acc[v] = X6[mb + v + moff][nb + l15];
    for (int kb = 0; kb < 48; kb += 4) {
      v2f a, b;
      a.x = S[mb + l15][kb + koff];
      a.y = S[mb + l15][kb + koff + 1];
      b.x = X3[kb + koff][nb + l15];
      b.y = X3[kb + koff + 1][nb + l15];
      acc = wmma4(a, b, acc);
    }
    for (int v = 0; v < 8; ++v) X6[mb + v + moff][nb + l15] = acc[v];
  }
  __syncthreads();

  // ---- score4 = softmax_k(x6 . center), out = sum_k score4 * x6 ----
  if (tid < 36) {
    const int r = tid, t = r / 9;
    float dot = 0.f;
    for (int c = 0; c < 64; ++c) dot += X6[r][c] * X6[t * 9 + 4][c];
    s4[r] = dot;
  }
  __syncthreads();
  if (tid < 4) {
    const int t = tid;
    float mx = s4[t * 9];
    for (int k = 1; k < 9; ++k) mx = fmaxf(mx, s4[t * 9 + k]);
    float z = 0.f, e[9];
    for (int k = 0; k < 9; ++k) { e[k] = __expf(s4[t * 9 + k] - mx); z += e[k]; }
    float inv = 1.0f / z;
    for (int k = 0; k < 9; ++k) s4[t * 9 + k] = e[k] * inv;
  }
  __syncthreads();

  float* ob = out + (size_t)nIdx * TT * CC * HWP + p;
  for (int e = tid; e < 256; e += 128) {
    int t = e >> 6, c = e & 63;
    float acc = 0.f;
    for (int k = 0; k < 9; ++k) acc += X6[t * 9 + k][c] * s4[t * 9 + k];
    ob[(size_t)(t * CC + c) * HWP] = acc;
  }
}

extern "C" void kernel_launch(void* const* d_in, const int* in_sizes, int n_in,
                              void* d_out, int out_size, void* d_ws, size_t ws_size,
                              hipStream_t stream) {
  const float* x  = (const float*)d_in[0];
  const float* w1 = (const float*)d_in[1];
  const float* w2 = (const float*)d_in[2];
  const float* w3 = (const float*)d_in[3];
  const int*  dil = (const int*)d_in[4];

  float* y = (float*)d_ws;  // 3 * YSZ * 4B = 56.6 MB of conv intermediates (L2-resident)

  dim3 gA(HWP / 32, NIMG);
  conv1x1_relu3<<<gA, 256, 0, stream>>>(x, w1, w2, w3, y);
  mgl_attn<<<2 * HWP, 128, 0, stream>>>(y, (float*)d_out, dil);

  (void)in_sizes; (void)n_in; (void)out_size; (void)ws_size;
}